// MPSPeriodic_39840116637930
// MI455X (gfx1250) — compile-verified
//
#include <hip/hip_runtime.h>
#include <math.h>

// ---------------------------------------------------------------------------
// MPS periodic chain product:  out[b] = log( trace( prod_l kernel[l, x[b,l]] ) )
// B=4096 sequences, L=128 positions, 8x8 fp32 matrices.
//
// MI455X (gfx1250) strategy:
//  * order-preserving binary tree over the 128-matrix chain (log depth)
//  * each pair of 8x8 fp32 products packed block-diagonally into one 16x16
//    V_WMMA_F32_16X16X4_F32 pair (K=8 as two K=4 slices, C-accumulate)
//  * per-lane operand bases are loop-invariant; products only add slot offsets
//  * phase 2a: each wave32 privately reduces a 16-matrix chunk (no barriers,
//    LDS ops from one wave are in-order); phase 2b: 3 cross-wave levels
//  * one workgroup (8 waves) owns 2 sequences; 64 KB LDS staging
// ---------------------------------------------------------------------------

typedef __attribute__((ext_vector_type(2))) float v2f;
typedef __attribute__((ext_vector_type(8))) float v8f;

#define LPOS 128
#define BDIM 8
#define MATF 64                    // floats per 8x8 matrix
#define BATCH_STRIDE (LPOS * MATF) // 8192 floats of LDS per sequence

// Full-wave product:  slot[ls] = slot[ls] x slot[ls + rdelta]   (per batch,
// block-diagonal two-batch packing). aLane/bLane/dLane are per-lane constant
// float offsets (include the batch-b BATCH_STRIDE bias and the lane-half K
// base); ls/rdelta are uniform slot offsets. EXEC must be all ones on entry.
__device__ __forceinline__ void wave_pair_product(float* mats, int ls, int rdelta,
                                                  int aLane, int bLane, int dLane,
                                                  bool storeLane) {
  const float* Ap = &mats[aLane + ls];           // A row m, K = k0.. (16x4 layout)
  const float* Bp = &mats[bLane + ls + rdelta];  // B col m, K = k0.. (4x16 layout)
  v2f a0, a1, b0, b1;
  a0.x = Ap[0];  a0.y = Ap[1];                   // K slice 0: k0, k0+1
  a1.x = Ap[4];  a1.y = Ap[5];                   // K slice 1: k0+4, k0+5
  b0.x = Bp[0];  b0.y = Bp[8];                   // rows k0, k0+1 of right matrix
  b1.x = Bp[32]; b1.y = Bp[40];                  // rows k0+4, k0+5

  v8f c = {};
  c = __builtin_amdgcn_wmma_f32_16x16x4_f32(false, a0, false, b0,
                                            (short)0, c, false, false);
  c = __builtin_amdgcn_wmma_f32_16x16x4_f32(false, a1, false, b1,
                                            (short)0, c, false, false);

  // D layout: VGPR v, lane l -> element (v + 8*(l>>4), l&15). Diagonal-block
  // lanes: 0-7 (batch a columns) and 24-31 (batch b columns).
  if (storeLane) {
    float* Dp = &mats[dLane + ls];
#pragma unroll
    for (int v = 0; v < 8; ++v) Dp[v * BDIM] = c[v];
  }
}

__global__ __launch_bounds__(256) void
mps_periodic_kernel(const int* __restrict__ x, const float* __restrict__ kern,
                    float* __restrict__ out, int Bc) {
  __shared__ float mats[2 * BATCH_STRIDE]; // 64 KB: two sequences' matrices

  const int bid = blockIdx.x;
  const int b0  = bid * 2;

  // ---- Phase 1: gather selected 8x8 matrices into LDS (float4 chunks) ------
  // 2 seq * 128 pos * 16 float4-chunks = 4096 chunks; consecutive threads take
  // consecutive chunks -> coalesced global_load_b128 / conflict-free b128 LDS.
  const float4* k4 = reinterpret_cast<const float4*>(kern);
  for (int idx = threadIdx.x; idx < 2 * LPOS * 16; idx += 256) {
    const int p    = idx >> 11;
    const int rem  = idx & 2047;
    const int l    = rem >> 4;
    const int e4   = rem & 15;
    const int phys = x[(b0 + p) * LPOS + l];
    const float4 val = k4[(l * 2 + phys) * 16 + e4];
    *reinterpret_cast<float4*>(&mats[p * BATCH_STRIDE + l * MATF + e4 * 4]) = val;
  }
  __syncthreads();

  const int lane = (int)(threadIdx.x & 31u);
  const int wave = (int)(threadIdx.x >> 5);

  // Loop-invariant per-lane operand bases (batch select folded in once).
  const int m  = lane & 15;
  const int k0 = (lane >> 4) << 1;
  const int aLane = ((m < 8) ? m * BDIM : BATCH_STRIDE + (m - 8) * BDIM) + k0;
  const int bLane = ((m < 8) ? m        : BATCH_STRIDE + (m - 8)) + k0 * BDIM;
  const bool storeLane = (lane < 8) || (lane >= 24);
  const int dLane = (lane < 8) ? lane : BATCH_STRIDE + (lane - 24);

  // ---- Phase 2a: private in-wave reduction of a 16-matrix chunk ------------
  // LDS ops from the same wave complete in order -> no barriers needed.
  {
    const int cb = wave * 16 * MATF;
    for (int stride = 1; stride < 16; stride <<= 1)
      for (int j = 0; j < 8 / stride; ++j)
        wave_pair_product(mats, cb + 2 * j * stride * MATF, stride * MATF,
                          aLane, bLane, dLane, storeLane);
  }
  __syncthreads();

  // ---- Phase 2b: cross-wave tree levels (stride 16, 32, 64) ----------------
  for (int stride = 16; stride < LPOS; stride <<= 1) {
    const int nprod = LPOS / (2 * stride);
    if (wave < nprod)
      wave_pair_product(mats, 2 * wave * stride * MATF, stride * MATF,
                        aLane, bLane, dLane, storeLane);
    __syncthreads();
  }

  // ---- Phase 3: trace + log ------------------------------------------------
  if (threadIdx.x < 2) {
    const int p = (int)threadIdx.x;
    if (b0 + p < Bc) {
      float t = 0.0f;
#pragma unroll
      for (int i = 0; i < BDIM; ++i)
        t += mats[p * BATCH_STRIDE + i * BDIM + i];
      out[b0 + p] = logf(t);
    }
  }
}

extern "C" void kernel_launch(void* const* d_in, const int* in_sizes, int n_in,
                              void* d_out, int out_size, void* d_ws, size_t ws_size,
                              hipStream_t stream) {
  const int*   x    = (const int*)d_in[0];   // (B, 128) int32
  const float* kern = (const float*)d_in[1]; // (128, 2, 8, 8) float32
  float*       out  = (float*)d_out;         // (B,) float32

  const int Bc     = in_sizes[0] / LPOS;     // 4096
  const int blocks = (Bc + 1) / 2;           // 2 sequences per workgroup

  mps_periodic_kernel<<<blocks, 256, 0, stream>>>(x, kern, out, Bc);
}